// FlatMixedEffectsDecoder_34282428957025
// MI455X (gfx1250) — compile-verified
//
#include <hip/hip_runtime.h>
#include <hip/hip_bf16.h>

// Problem constants (match reference)
#define B_SZ 256
#define P_SZ 64
#define M_SZ 500
#define R_SZ 8
#define V_SZ 32000
#define RV_SZ (R_SZ * V_SZ)   // 256000

typedef __attribute__((ext_vector_type(2))) float v2f;
typedef __attribute__((ext_vector_type(8))) float v8f;

// ---------------------------------------------------------------------------
// Kernel 1: bucket batch rows by role (compacted index lists in workspace).
// Single block; re-initializes counts every call (idempotent across replays).
// ---------------------------------------------------------------------------
__global__ void build_role_lists(const int* __restrict__ r_idx,
                                 int* __restrict__ counts,   // [R]
                                 int* __restrict__ rows) {   // [R][B]
  const int t = threadIdx.x;
  if (t < R_SZ) counts[t] = 0;
  __syncthreads();
  if (t < B_SZ) {
    const int r = r_idx[t];
    const int pos = atomicAdd(&counts[r], 1);
    rows[r * B_SZ + pos] = t;
  }
}

// ---------------------------------------------------------------------------
// Kernel 2: per (role, 64-col slab) block. 4 waves; each wave owns a 16-col
// N-subtile. B-fragment (64x16 f32 persona tile) is loaded ONCE per wave and
// reused across all M-tiles of this role -> eta_persona streamed exactly once.
// GEMM via 16 chained V_WMMA_F32_16X16X4_F32 (K=64). Branch-free inner loop:
// clamped indices + v_cndmask selects instead of exec-mask predication;
// per-element row/author indices obtained with __shfl (no global re-gather).
// ---------------------------------------------------------------------------
__global__ void wmma_logits(const int*   __restrict__ m_idx,        // [B]
                            const float* __restrict__ z,            // [B][P]
                            const int*   __restrict__ role_mask,    // [R][V]
                            const float* __restrict__ eta_bg,       // [R][V]
                            const float* __restrict__ eta_author,   // [M][R][V]
                            const float* __restrict__ eta_persona,  // [P][R][V]
                            const int*   __restrict__ counts,
                            const int*   __restrict__ rows,
                            float*       __restrict__ out) {        // [B][V]
  const int r = blockIdx.y;
  const int nrows = counts[r];           // block-uniform
  if (nrows == 0) return;                // wave-uniform exit: EXEC stays full

  const int wave = threadIdx.x >> 5;     // 0..3
  const int lane = threadIdx.x & 31;
  const int hi   = lane >> 4;            // half-wave selector (K/M offset)
  const int nsub = lane & 15;            // N (and A-row M) index within tile
  const int col  = blockIdx.x * 64 + wave * 16 + nsub;

  // --- Preload B fragment: 4x16 f32 per WMMA step, 16 steps covering K=64.
  // v0 holds K=k0+2*hi, v1 holds K+1; lanes 0-15 / 16-31 both carry N=0..15.
  const float* Wp = eta_persona + (size_t)r * V_SZ;  // elem: p*RV + col
  v2f bfrag[16];
#pragma unroll
  for (int i = 0; i < 16; ++i) {
    const int k = 4 * i + 2 * hi;
    bfrag[i].x = Wp[(size_t)k * RV_SZ + col];
    bfrag[i].y = Wp[(size_t)(k + 1) * RV_SZ + col];
  }

  const float bgv = eta_bg[(size_t)r * V_SZ + col];
  const int   mk  = role_mask[(size_t)r * V_SZ + col];
  const int*  rl  = rows + r * B_SZ;

  const int mt_count = (nrows + 15) >> 4;
  for (int mt = 0; mt < mt_count; ++mt) {
    // --- A fragment: 16x4 f32 tile rows gathered via compacted role list.
    // Clamp the list index so every load address is valid; zero pad rows with
    // selects (v_cndmask) rather than divergent exec-masked loads.
    const int  am     = mt * 16 + nsub;
    const int  amc    = (am < nrows) ? am : (nrows - 1);
    const bool avalid = (am < nrows);
    const int  arow   = rl[amc];                 // this lane's batch row
    const int  mrow   = m_idx[arow];             // its author index (1 load)
    const float* zp   = z + arow * P_SZ;

    v2f afrag[16];
#pragma unroll
    for (int i = 0; i < 16; ++i) {
      const int k = 4 * i + 2 * hi;              // 8B-aligned (k even)
      const v2f t = *(const v2f*)(zp + k);
      afrag[i].x = avalid ? t.x : 0.0f;
      afrag[i].y = avalid ? t.y : 0.0f;
    }

    // --- 16 chained f32 WMMAs: D = A(16x4) * B(4x16) + C
    v8f acc = {};
#pragma unroll
    for (int i = 0; i < 16; ++i) {
      acc = __builtin_amdgcn_wmma_f32_16x16x4_f32(
          /*neg_a=*/false, afrag[i],
          /*neg_b=*/false, bfrag[i],
          /*c_mod=*/(short)0, acc,
          /*reuse_a=*/false, /*reuse_b=*/false);
    }

    // --- Fused epilogue. C/D layout: VGPR i, lanes 0-15 -> M=i, N=lane;
    // lanes 16-31 -> M=i+8, N=lane-16. Row/author indices for element i live
    // in lane (i + 8*hi): fetch with cross-lane shuffle, no global re-gather.
#pragma unroll
    for (int i = 0; i < 8; ++i) {
      const int src  = i + 8 * hi;                       // source lane 0..15
      const int gb_i = __shfl(arow, src, 32);
      const int m_i  = __shfl(mrow, src, 32);
      const bool valid = (mt * 16 + src) < nrows;
      // Unconditional gather (address safe via clamped rows); mask via select.
      const float auth =
          eta_author[((size_t)m_i * R_SZ + r) * V_SZ + col];
      const float logit = (mk != 0) ? (acc[i] + bgv + auth) : -100.0f;
      if (valid) out[(size_t)gb_i * V_SZ + col] = logit;
    }
  }
}

// ---------------------------------------------------------------------------
// Kernel 3: in-place row-wise log-softmax. One block per row; deterministic
// strided float4 sum + LDS tree reduction (no cross-block float atomics).
// Logits are bounded (|x| ~ 1, mask fill = -100) so no max-shift is needed
// for fp32 stability.
// ---------------------------------------------------------------------------
__global__ void row_log_softmax(float* __restrict__ out) {
  __shared__ float red[256];
  __shared__ float s_logZ;
  const int b = blockIdx.x;
  float4* rowp = (float4*)(out + (size_t)b * V_SZ);   // 16B-aligned row
  const int nv4 = V_SZ / 4;                           // 8000

  float s = 0.0f;
  for (int v = threadIdx.x; v < nv4; v += 256) {
    const float4 x = rowp[v];
    s += expf(x.x) + expf(x.y) + expf(x.z) + expf(x.w);
  }
  red[threadIdx.x] = s;
  __syncthreads();
  for (int off = 128; off > 0; off >>= 1) {
    if (threadIdx.x < off) red[threadIdx.x] += red[threadIdx.x + off];
    __syncthreads();
  }
  if (threadIdx.x == 0) s_logZ = logf(red[0]);
  __syncthreads();
  const float lz = s_logZ;
  for (int v = threadIdx.x; v < nv4; v += 256) {
    float4 x = rowp[v];
    x.x -= lz; x.y -= lz; x.z -= lz; x.w -= lz;
    rowp[v] = x;
  }
}

// ---------------------------------------------------------------------------
extern "C" void kernel_launch(void* const* d_in, const int* in_sizes, int n_in,
                              void* d_out, int out_size, void* d_ws,
                              size_t ws_size, hipStream_t stream) {
  const int*   m_idx       = (const int*)d_in[0];    // (B,)
  const float* z_persona   = (const float*)d_in[1];  // (B,P)
  const int*   r_idx       = (const int*)d_in[2];    // (B,)
  const int*   role_mask   = (const int*)d_in[3];    // (R,V)
  const float* eta_bg      = (const float*)d_in[4];  // (R,V)
  const float* eta_author  = (const float*)d_in[5];  // (M,R,V)
  const float* eta_persona = (const float*)d_in[6];  // (P,R,V)
  float* out = (float*)d_out;                        // (B,V)

  int* counts = (int*)d_ws;          // 8 ints
  int* rows   = counts + R_SZ;       // 8*256 ints  (total 8.2 KB of ws)

  build_role_lists<<<1, 256, 0, stream>>>(r_idx, counts, rows);

  dim3 grid(V_SZ / 64, R_SZ);        // (500, 8)
  wmma_logits<<<grid, 128, 0, stream>>>(m_idx, z_persona, role_mask, eta_bg,
                                        eta_author, eta_persona, counts, rows,
                                        out);

  row_log_softmax<<<B_SZ, 256, 0, stream>>>(out);
}